// ConeProjection_68642167325009
// MI455X (gfx1250) — compile-verified
//
#include <hip/hip_runtime.h>

typedef __attribute__((ext_vector_type(2))) float v2f;
typedef __attribute__((ext_vector_type(8))) float v8f;

#define NGRID  169
#define NTILES 11

// ---------------------------------------------------------------------------
// Compile-time table of per-lane WMMA B-fragments for Q[8,169].
// Q rows: q0=x^2 q1=y^2 q2=xy q3=x q4=y q5=1 (q6=q7=0), x=((n/13)-6)/6,
// y=((n%13)-6)/6.  B layout (4x16 f32): VGPR0 = K0 (lanes 0-15) / K2 (16-31);
// VGPR1 = K1 / K3.  Entry [tile][lane] = {b0.x, b0.y, b1.x, b1.y}.
// Columns n >= 169 hold dummy values (their outputs are never stored).
// ---------------------------------------------------------------------------
struct alignas(16) QTab { float q[NTILES][32][4]; };

static constexpr QTab make_qtab() {
    QTab tq{};
    for (int tile = 0; tile < NTILES; ++tile) {
        for (int lane = 0; lane < 32; ++lane) {
            const int  m  = lane & 15;
            const bool hi = lane >= 16;
            int n = tile * 16 + m;
            if (n >= NGRID) n = 0;
            const float x = (float)((n / 13) - 6) / 6.0f;
            const float y = (float)((n % 13) - 6) / 6.0f;
            tq.q[tile][lane][0] = hi ? x * y : x * x;  // K2 : K0
            tq.q[tile][lane][1] = hi ? x     : y * y;  // K3 : K1
            tq.q[tile][lane][2] = hi ? 0.f   : y;      // K6 : K4
            tq.q[tile][lane][3] = hi ? 0.f   : 1.f;    // K7 : K5
        }
    }
    return tq;
}
__constant__ QTab g_qtab = make_qtab();

// One 16x16 output tile via two V_WMMA_F32_16X16X4_F32 (K padded 6->8).
__device__ __forceinline__ v8f cone_tile_mm(v2f a0, v2f a1, float4 qv) {
    v2f b0; b0.x = qv.x; b0.y = qv.y;   // K = 0..3
    v2f b1; b1.x = qv.z; b1.y = qv.w;   // K = 4..7
    v8f acc = {};
    acc = __builtin_amdgcn_wmma_f32_16x16x4_f32(false, a0, false, b0,
                                                (short)0, acc, false, false);
    acc = __builtin_amdgcn_wmma_f32_16x16x4_f32(false, a1, false, b1,
                                                (short)0, acc, false, false);
    return acc;
}

// out[B,169] = C[B,8] x Q[8,169]   (f32 end-to-end)
// Each wave: 16 batches (M) x 169 grid points (N) as 11 N-tiles of 16.
__global__ __launch_bounds__(256) void ConeProjection_wmma_kernel(
    const float* __restrict__ eyes,   // [B,3]
    const float* __restrict__ v,      // [B,3]
    const float* __restrict__ R,      // [B,3,3] row-major
    const float* __restrict__ t,      // [B,3]
    const float* __restrict__ alpha,  // [B]
    float* __restrict__ out,          // [B,169]
    int Btot)
{
    const int lane = threadIdx.x & 31;
    const int wave = threadIdx.x >> 5;
    const int m    = lane & 15;
    const bool hi  = lane >= 16;

    const int tileBase = (blockIdx.x * 8 + wave) * 16;  // first batch of this wave's tile
    int b = tileBase + m;                                // lanes 16..31 recompute batch (lane-16)
    if (b >= Btot) b = Btot - 1;                         // clamp loads (no EXEC divergence)

    // ---- per-batch coefficients (redundant x2 across lane halves, ~70 flops) ----
    const float vx = v[3*b+0], vy = v[3*b+1], vz = v[3*b+2];
    const float inv = 1.0f / fmaxf(sqrtf(vx*vx + vy*vy + vz*vz), 1e-14f);
    const float nx = vx*inv, ny = vy*inv, nz = vz*inv;

    const float* Rb = R + 9*b;
    const float w0x = Rb[0], w0y = Rb[3], w0z = Rb[6];   // R[:,0]
    const float w1x = Rb[1], w1y = Rb[4], w1z = Rb[7];   // R[:,1]
    const float w2x = t[3*b+0]-eyes[3*b+0];
    const float w2y = t[3*b+1]-eyes[3*b+1];
    const float w2z = t[3*b+2]-eyes[3*b+2];
    const float al = alpha[b];

    const float d0 = nx*w0x + ny*w0y + nz*w0z;
    const float d1 = nx*w1x + ny*w1y + nz*w1z;
    const float d2 = nx*w2x + ny*w2y + nz*w2z;

    const float s00 = d0*d0 - al*(w0x*w0x + w0y*w0y + w0z*w0z);
    const float s11 = d1*d1 - al*(w1x*w1x + w1y*w1y + w1z*w1z);
    const float s22 = d2*d2 - al*(w2x*w2x + w2y*w2y + w2z*w2z);
    const float s01 = d0*d1 - al*(w0x*w1x + w0y*w1y + w0z*w1z);
    const float s02 = d0*d2 - al*(w0x*w2x + w0y*w2y + w0z*w2z);
    const float s12 = d1*d2 - al*(w1x*w2x + w1y*w2y + w1z*w2z);

    // out[k] = c0*x^2 + c1*y^2 + c2*x*y + c3*x + c4*y + c5   (c6=c7=0 pad)
    const float c0 = s00, c1 = s11, c2 = 2.f*s01, c3 = 2.f*s02, c4 = 2.f*s12, c5 = s22;

    // A layout (16x4 f32): VGPR0 = K0 (lanes 0-15) / K2 (16-31); VGPR1 = K1 / K3.
    v2f a0; a0.x = hi ? c2 : c0;  a0.y = hi ? c3  : c1;   // K = 0..3
    v2f a1; a1.x = hi ? 0.f : c4; a1.y = hi ? 0.f : c5;   // K = 4..7 (rows 6,7 zero)

    // Per-lane B-fragment table pointer and output base.
    const float4* __restrict__ qt = (const float4*)g_qtab.q;   // [NTILES][32]
    float* __restrict__ po = out + (long)(tileBase + (hi ? 8 : 0)) * NGRID + m;

    if (tileBase + 16 <= Btot) {
        // ---- fast path: tiles 0..9 fully valid, no guards at all ----
        #pragma unroll
        for (int tile = 0; tile < 10; ++tile) {
            v8f acc = cone_tile_mm(a0, a1, qt[tile * 32 + lane]);
            const int n = tile * 16 + m;
            #pragma unroll
            for (int r = 0; r < 8; ++r)
                po[(long)r * NGRID + n] = acc[r];      // straight-line store clause
        }
        // tail tile: columns 160..168 valid (m < 9), single exec mask on stores
        {
            v8f acc = cone_tile_mm(a0, a1, qt[10 * 32 + lane]);
            const int n = 160 + m;
            if (m < 9) {
                #pragma unroll
                for (int r = 0; r < 8; ++r)
                    po[(long)r * NGRID + n] = acc[r];
            }
        }
    } else {
        // ---- rare partial-batch tail (never taken when Btot % 128 == 0) ----
        const int rowLimit = Btot - tileBase;
        for (int tile = 0; tile < NTILES; ++tile) {
            v8f acc = cone_tile_mm(a0, a1, qt[tile * 32 + lane]);
            const int n = tile * 16 + m;
            if (n < NGRID) {
                #pragma unroll
                for (int r = 0; r < 8; ++r) {
                    const int r16 = (hi ? 8 : 0) + r;
                    if (r16 < rowLimit)
                        po[(long)r * NGRID + n] = acc[r];
                }
            }
        }
    }
}

extern "C" void kernel_launch(void* const* d_in, const int* in_sizes, int n_in,
                              void* d_out, int out_size, void* d_ws, size_t ws_size,
                              hipStream_t stream) {
    const float* eyes  = (const float*)d_in[0];
    const float* v     = (const float*)d_in[1];
    const float* R     = (const float*)d_in[2];
    const float* t     = (const float*)d_in[3];
    const float* alpha = (const float*)d_in[4];
    float* out = (float*)d_out;

    const int Btot   = in_sizes[4];          // alpha is [B]
    const int blocks = (Btot + 127) / 128;   // 128 batches/block = 8 waves x 16
    ConeProjection_wmma_kernel<<<blocks, 256, 0, stream>>>(eyes, v, R, t, alpha, out, Btot);
}